// ProdLayer_43069932044330
// MI455X (gfx1250) — compile-verified
//
#include <hip/hip_runtime.h>
#include <stdint.h>

// Sentinel for "empty segment": a signaling-NaN bit pattern that cannot be
// produced by v_mul_f32 (hardware NaN results are quieted/canonical).
#define SENT 0x7F800001u

constexpr int BLOCK = 256;
constexpr int CHUNK = 16;                 // edges per thread
constexpr int TILE  = BLOCK * CHUNK;      // 4096 edges per block

// ---------------------------------------------------------------------------
// CAS-based atomic multiply that treats the SENT bit pattern as identity (1.0)
// ---------------------------------------------------------------------------
__device__ __forceinline__ void atomic_mul_sent(float* addr, float val) {
    unsigned int* ua = (unsigned int*)addr;
    unsigned int cur = __hip_atomic_load(ua, __ATOMIC_RELAXED,
                                         __HIP_MEMORY_SCOPE_AGENT);
    while (true) {
        unsigned int assumed = cur;
        float base = (assumed == SENT) ? 1.0f : __uint_as_float(assumed);
        cur = atomicCAS(ua, assumed, __float_as_uint(base * val));
        if (cur == assumed) break;
    }
}

// ---------------------------------------------------------------------------
// Kernel 1: fill out[] with the sentinel (vectorized b128 stores).
// out is re-read by seg_prod/seg_fix -> default (RT) caching is correct.
// ---------------------------------------------------------------------------
__global__ void seg_init(unsigned int* __restrict__ out, int S) {
    int i  = blockIdx.x * blockDim.x + threadIdx.x;
    int i4 = i * 4;
    if (i4 + 3 < S) {
        ((uint4*)out)[i] = make_uint4(SENT, SENT, SENT, SENT);
    } else if (i4 < S) {
        for (int k = i4; k < S; ++k) out[k] = SENT;
    }
}

// ---------------------------------------------------------------------------
// Kernel 2: gather + segmented product.
// Each thread owns CHUNK contiguous edges; csr/ptrs staged to LDS via the
// CDNA5 async data mover (ASYNCcnt path) with non-temporal hint so the
// 128MB one-shot streams don't evict the L2-resident x (16MB) and out (16MB).
// ---------------------------------------------------------------------------
__global__ void __launch_bounds__(BLOCK)
seg_prod(const float* __restrict__ x, const int* __restrict__ ptrs,
         const int* __restrict__ csr, float* __restrict__ out, int E) {
    __shared__ __align__(16) int s_csr[TILE];
    __shared__ __align__(16) int s_ptr[TILE];

    const int t = threadIdx.x;
    const long long gbase = (long long)blockIdx.x * TILE + (long long)t * CHUNK;
    if (gbase >= E) return;
    const long long rem = (long long)E - gbase;
    const int n = rem >= CHUNK ? CHUNK : (int)rem;

    if (n == CHUNK) {
        // --- async stage: 64B of csr + 64B of ptrs per thread -> LDS -------
        // Generic LDS address low 32 bits == LDS byte offset (ISA 10.2).
        unsigned int lds_c = (unsigned int)(uintptr_t)(&s_csr[t * CHUNK]);
        unsigned int lds_p = (unsigned int)(uintptr_t)(&s_ptr[t * CHUNK]);
        unsigned long long gc = (unsigned long long)(uintptr_t)(csr  + gbase);
        unsigned long long gp = (unsigned long long)(uintptr_t)(ptrs + gbase);
#pragma unroll
        for (int i = 0; i < 4; ++i) {
            asm volatile("global_load_async_to_lds_b128 %0, %1, off th:TH_LOAD_NT"
                         :: "v"(lds_c + i * 16), "v"(gc + (unsigned long long)(i * 16))
                         : "memory");
            asm volatile("global_load_async_to_lds_b128 %0, %1, off th:TH_LOAD_NT"
                         :: "v"(lds_p + i * 16), "v"(gp + (unsigned long long)(i * 16))
                         : "memory");
        }
        asm volatile("s_wait_asynccnt 0x0" ::: "memory");

        // --- pull tile back as b128 LDS loads ------------------------------
        int c[CHUNK];
        int p[CHUNK];
        const int4* c4 = (const int4*)&s_csr[t * CHUNK];
        const int4* p4 = (const int4*)&s_ptr[t * CHUNK];
#pragma unroll
        for (int i = 0; i < 4; ++i) {
            int4 cc = c4[i]; int4 pp = p4[i];
            c[4*i+0] = cc.x; c[4*i+1] = cc.y; c[4*i+2] = cc.z; c[4*i+3] = cc.w;
            p[4*i+0] = pp.x; p[4*i+1] = pp.y; p[4*i+2] = pp.z; p[4*i+3] = pp.w;
        }

        // --- issue all gathers up front (L2-resident: x is 16MB << 192MB) --
        float v[CHUNK];
#pragma unroll
        for (int j = 0; j < CHUNK; ++j) v[j] = x[p[j]];

        // --- run-length flush: interior runs are exclusively owned ---------
        int   seg   = c[0];
        float prod  = v[0];
        bool  first = true;
#pragma unroll
        for (int j = 1; j < CHUNK; ++j) {
            if (c[j] != seg) {
                if (first) { atomic_mul_sent(out + seg, prod); first = false; }
                else       { out[seg] = prod; }  // exclusive: starts+ends here
                seg  = c[j];
                prod = v[j];
            } else {
                prod *= v[j];
            }
        }
        atomic_mul_sent(out + seg, prod);        // may continue into next chunk
    } else {
        // --- rare tail: plain scalar path (streams are one-shot -> NT) -----
        int   seg   = __builtin_nontemporal_load(csr + gbase);
        float prod  = x[__builtin_nontemporal_load(ptrs + gbase)];
        bool  first = true;
        for (int i = 1; i < n; ++i) {
            int   cc = __builtin_nontemporal_load(csr + gbase + i);
            float vv = x[__builtin_nontemporal_load(ptrs + gbase + i)];
            if (cc != seg) {
                if (first) { atomic_mul_sent(out + seg, prod); first = false; }
                else       { out[seg] = prod; }
                seg = cc; prod = vv;
            } else {
                prod *= vv;
            }
        }
        atomic_mul_sent(out + seg, prod);
    }
}

// ---------------------------------------------------------------------------
// Kernel 3: sentinel (empty segment) -> 0.0f
// ---------------------------------------------------------------------------
__global__ void seg_fix(float* __restrict__ out, int S) {
    int i  = blockIdx.x * blockDim.x + threadIdx.x;
    int i4 = i * 4;
    if (i4 + 3 < S) {
        uint4 u = ((const uint4*)out)[i];
        float4 f;
        f.x = (u.x == SENT) ? 0.0f : __uint_as_float(u.x);
        f.y = (u.y == SENT) ? 0.0f : __uint_as_float(u.y);
        f.z = (u.z == SENT) ? 0.0f : __uint_as_float(u.z);
        f.w = (u.w == SENT) ? 0.0f : __uint_as_float(u.w);
        ((float4*)out)[i] = f;
    } else if (i4 < S) {
        for (int k = i4; k < S; ++k) {
            unsigned int u = ((const unsigned int*)out)[k];
            out[k] = (u == SENT) ? 0.0f : __uint_as_float(u);
        }
    }
}

// ---------------------------------------------------------------------------
extern "C" void kernel_launch(void* const* d_in, const int* in_sizes, int n_in,
                              void* d_out, int out_size, void* d_ws, size_t ws_size,
                              hipStream_t stream) {
    const float* x    = (const float*)d_in[0];
    const int*   ptrs = (const int*)d_in[1];
    const int*   csr  = (const int*)d_in[2];
    float*       out  = (float*)d_out;

    const int E = in_sizes[1];   // edges
    const int S = out_size;      // segments

    const int s4      = (S + 3) / 4;
    const int initGrd = (s4 + BLOCK - 1) / BLOCK;
    const int mainGrd = (int)(((long long)E + TILE - 1) / TILE);

    seg_init<<<initGrd, BLOCK, 0, stream>>>((unsigned int*)out, S);
    seg_prod<<<mainGrd, BLOCK, 0, stream>>>(x, ptrs, csr, out, E);
    seg_fix <<<initGrd, BLOCK, 0, stream>>>(out, S);
}